// Casual_Conv1D_69604239999093
// MI455X (gfx1250) — compile-verified
//
#include <hip/hip_runtime.h>
#include <hip/hip_bf16.h>

typedef __attribute__((ext_vector_type(2))) float v2f;
typedef __attribute__((ext_vector_type(8))) float v8f;

#define BATCH      128
#define LEN        16384
#define NLAYERS    31
#define OUTL       (LEN - NLAYERS)      // 16353
#define TILE       256                  // outputs per wave (16x16 WMMA tile group)
#define TILES_ROW  64                   // ceil(16353/256)
#define WAVES_BLK  8                    // 256 threads, wave32

// d_ws layout (floats):
//   [0..31]    composed FIR kernel c
//   [32]       composed bias d
//   [64..831]  per-lane Toeplitz A-fragment table: lane*24 + 2p + {0,1}
//   [832..863] per-lane dump slots for predicated tail stores
#define WS_ATAB    64
#define WS_DUMP    832

// ---------------------------------------------------------------------------
// Kernel 1 (32 threads): compose the 31 two-tap kernels into one 32-tap FIR
// plus bias scalar, then build the wave-invariant Toeplitz A-fragment table.
// ---------------------------------------------------------------------------
__global__ void compose_fir_kernel(const float* __restrict__ W,
                                   const float* __restrict__ b,
                                   float* __restrict__ ws) {
    __shared__ float sc[32];
    const int t = threadIdx.x;

    if (t == 0) {
        float c[32];
        c[0] = 1.0f;
        int len = 1;
        float d = 0.0f;
        for (int i = 0; i < NLAYERS; ++i) {
            const float w0 = W[2 * i + 0];
            const float w1 = W[2 * i + 1];
            float nc[32];
            for (int k = 0; k <= len; ++k) {
                const float a  = (k < len) ? c[k]     : 0.0f;
                const float bb = (k >= 1)  ? c[k - 1] : 0.0f;
                nc[k] = w0 * a + w1 * bb;
            }
            ++len;
            for (int k = 0; k < len; ++k) c[k] = nc[k];
            d = (w0 + w1) * d + b[i];
        }
        for (int k = 0; k < 32; ++k) { ws[k] = c[k]; sc[k] = c[k]; }
        ws[32] = d;
    }
    __syncthreads();

    // A-fragment p covers K = 4p..4p+3; lane holds (M=n, K=4p+2h, 4p+2h+1)
    const int h = t >> 4;
    const int n = t & 15;
    for (int p = 0; p < 12; ++p) {
        const int k0 = 4 * p + 2 * h - n;
        const int k1 = k0 + 1;
        ws[WS_ATAB + t * 24 + 2 * p + 0] = (k0 >= 0 && k0 < 32) ? sc[k0] : 0.0f;
        ws[WS_ATAB + t * 24 + 2 * p + 1] = (k1 >= 0 && k1 < 32) ? sc[k1] : 0.0f;
    }
}

// ---------------------------------------------------------------------------
// Kernel 2: one wave computes 256 contiguous outputs of one batch row as a
// 16x16 fp32 tile via 12 chained V_WMMA_F32_16X16X4_F32 (K = 0..47 banded
// Toeplitz of the composed FIR kernel).
//   Y[M][N] = sum_k c[k-M] * x[tbase + 16*N + k]  ->  out[tbase + 16*N + M]
// ---------------------------------------------------------------------------
__global__ __launch_bounds__(256) void fir_wmma_kernel(const float* __restrict__ x,
                                                       float* __restrict__ ws,
                                                       float* __restrict__ out) {
    __shared__ float x_lds[WAVES_BLK][320];   // 288 used, padded

    const int tid  = threadIdx.x;
    const int lane = tid & 31;
    const int wv   = tid >> 5;
    const int h    = lane >> 4;               // lane half (selects K parity pair)
    const int n    = lane & 15;               // N column / M row index

    const float bias = ws[32];

    const int  tile  = blockIdx.x * WAVES_BLK + wv;   // 0 .. 8191
    const int  brow  = tile >> 6;                     // / TILES_ROW
    const int  tIdx  = tile & (TILES_ROW - 1);
    const int  tbase = tIdx << 8;                     // * TILE
    const long xoff  = (long)brow * LEN;
    const long ooff  = (long)brow * OUTL;

    const bool full = (tIdx != TILES_ROW - 1);        // wave-uniform fast path

    // --- stage the 288-float input window into this wave's LDS region ------
    if (full) {
        // fully in-bounds: vector loads/stores, no guards
        const float4* xp = (const float4*)(x + xoff + tbase);   // 16B aligned
        const float4 a  = xp[lane];         // floats   0..127
        const float4 c4 = xp[lane + 32];    // floats 128..255
        const float  s  = x[xoff + tbase + 256 + lane];   // floats 256..287
        *(float4*)&x_lds[wv][4 * lane]       = a;
        *(float4*)&x_lds[wv][4 * lane + 128] = c4;
        x_lds[wv][256 + lane] = s;
    } else {
        // branchless tail: clamp address, select value to zero
#pragma unroll
        for (int q = 0; q < 9; ++q) {
            const int ix  = tbase + lane + 32 * q;
            const int ixc = (ix < LEN) ? ix : (LEN - 1);
            float v = x[xoff + ixc];
            v = (ix < LEN) ? v : 0.0f;
            x_lds[wv][lane + 32 * q] = v;
        }
    }

    // --- fetch this lane's 12 precomputed Toeplitz A-fragments (24 floats) -
    v2f A[12];
    const float4* at = (const float4*)(ws + WS_ATAB + lane * 24);  // 16B aligned
#pragma unroll
    for (int q = 0; q < 6; ++q) {
        const float4 f = at[q];
        A[2 * q + 0] = (v2f){f.x, f.y};
        A[2 * q + 1] = (v2f){f.z, f.w};
    }

    __syncthreads();   // x_lds visible across the wave's lanes

    // --- 12 chained fp32 WMMAs, accumulator pre-seeded with the bias -------
    v8f acc = {bias, bias, bias, bias, bias, bias, bias, bias};
    const float* xw = &x_lds[wv][0];
    const int nbase = 16 * n + 2 * h;     // float index; even -> 8B aligned
#pragma unroll
    for (int p = 0; p < 12; ++p) {
        const v2f Bv = *(const v2f*)(xw + nbase + 4 * p);   // X[4p+2h .. +1][n]
        acc = __builtin_amdgcn_wmma_f32_16x16x4_f32(
            /*neg_a=*/false, A[p], /*neg_b=*/false, Bv,
            /*c_mod=*/(short)0, acc, /*reuse_a=*/false, /*reuse_b=*/false);
    }

    // --- write back: acc[r] = Y[M=r+8h][N=n] -> out[tbase + 16n + r + 8h] --
    const int mofs = 16 * n + 8 * h;
    float* op = out + ooff + tbase + mofs;
    if (full) {
        // unguarded: compiler merges into 2x global_store_b128
#pragma unroll
        for (int r = 0; r < 8; ++r) op[r] = acc[r];
    } else {
        // branchless tail: steer out-of-range elements to a per-lane dump slot
        const int valid = OUTL - tbase;   // 225 for the tail tile
        float* dump = ws + WS_DUMP + lane;
#pragma unroll
        for (int r = 0; r < 8; ++r) {
            float* dst = (mofs + r < valid) ? (op + r) : dump;
            *dst = acc[r];
        }
    }
}

extern "C" void kernel_launch(void* const* d_in, const int* in_sizes, int n_in,
                              void* d_out, int out_size, void* d_ws, size_t ws_size,
                              hipStream_t stream) {
    const float* x = (const float*)d_in[0];   // (128, 1, 16384) fp32
    const float* W = (const float*)d_in[1];   // (31, 2) fp32
    const float* b = (const float*)d_in[2];   // (31,) fp32
    float* out = (float*)d_out;               // (128, 1, 16353) fp32
    float* ws  = (float*)d_ws;                // 864 floats used

    compose_fir_kernel<<<1, 32, 0, stream>>>(W, b, ws);

    const int total_tiles = BATCH * TILES_ROW;            // 8192 waves
    const int grid = total_tiles / WAVES_BLK;             // 1024 blocks
    fir_wmma_kernel<<<grid, 256, 0, stream>>>(x, ws, out);
}